// RotRandomizerForPrediction_69037304315954
// MI455X (gfx1250) — compile-verified
//
#include <hip/hip_runtime.h>
#include <hip/hip_bf16.h>
#include <math.h>

// ---------------------------------------------------------------------------
// RotRandomizerForPrediction for MI455X (gfx1250, wave32).
// Memory-bound: ~118 MB traffic -> ~5.1 us at 23.3 TB/s.  The per-point 3x3
// matmuls (R*rot, R*pos) are offloaded to V_WMMA_F32_16X16X4_F32 so they
// co-execute with the VALU quat<->matrix conversions (XDL ops track as TRANS).
// Round 2: hoist all 8 LDS A-operand loads ahead of the WMMA chain so the
// ds_load_b64s pipeline under a single s_wait_dscnt instead of serializing
// load->wait->wmma eight times.
// ---------------------------------------------------------------------------

#define BATCH 16384
#define TPTS  128

typedef __attribute__((ext_vector_type(2))) float v2f;
typedef __attribute__((ext_vector_type(8))) float v8f;

// ----------------------------- threefry2x32 --------------------------------
__device__ __forceinline__ unsigned rotl32(unsigned v, int n) {
    return (v << n) | (v >> (32 - n));
}

__device__ __forceinline__ void tf2x32(unsigned k0, unsigned k1,
                                       unsigned& x0, unsigned& x1) {
    const unsigned ks0 = k0, ks1 = k1, ks2 = k0 ^ k1 ^ 0x1BD11BDAu;
    x0 += ks0; x1 += ks1;
#define TFR(r) { x0 += x1; x1 = rotl32(x1, r); x1 ^= x0; }
    TFR(13) TFR(15) TFR(26) TFR(6)  x0 += ks1; x1 += ks2 + 1u;
    TFR(17) TFR(29) TFR(16) TFR(24) x0 += ks2; x1 += ks0 + 2u;
    TFR(13) TFR(15) TFR(26) TFR(6)  x0 += ks0; x1 += ks1 + 3u;
    TFR(17) TFR(29) TFR(16) TFR(24) x0 += ks1; x1 += ks2 + 4u;
    TFR(13) TFR(15) TFR(26) TFR(6)  x0 += ks2; x1 += ks0 + 5u;
#undef TFR
}

__device__ __forceinline__ float bits_to_unit(unsigned bits) {
    // JAX uniform: bitcast((bits>>9)|0x3f800000) - 1.0  in [0,1)
    return __uint_as_float((bits >> 9) | 0x3f800000u) - 1.0f;
}

// ---------------- Kernel 1: per-batch random rotation R --------------------
// JAX chain (threefry_partitionable): key(42) -> split -> sub -> split(4)
// -> uniform(k_i,(B,)).  Key derivation is all-constant: clang folds it.
__global__ __launch_bounds__(256) void
RotRandomizer_build_R_kernel(float* __restrict__ out_R) {
    const int b = blockIdx.x * 256 + threadIdx.x;

    // sub = split(key(42))[1]  : threefry(k=(0,42), counts=(0,1))
    unsigned s0 = 0u, s1 = 1u;
    tf2x32(0u, 42u, s0, s1);
    const unsigned sub0 = s0, sub1 = s1;

    // k[i] = split(sub, 4)[i] : threefry(sub, counts=(0,i))
    unsigned k[4][2];
#pragma unroll
    for (int i = 0; i < 4; ++i) {
        unsigned a = 0u, c = (unsigned)i;
        tf2x32(sub0, sub1, a, c);
        k[i][0] = a; k[i][1] = c;
    }

    // uniform draws for this batch element: bits = y0 ^ y1, counts=(0,b)
    float u[4];
#pragma unroll
    for (int i = 0; i < 4; ++i) {
        unsigned a = 0u, c = (unsigned)b;
        tf2x32(k[i][0], k[i][1], a, c);
        u[i] = bits_to_unit(a ^ c);
    }

    const float TWO_PI = 6.28318530717958647692f;
    const float PI_F   = 3.14159265358979323846f;
    float ax = u[0] * TWO_PI - PI_F;
    float ay = u[1] * TWO_PI - PI_F;
    float az = u[2] * TWO_PI - PI_F;
    if (u[3] < 0.015625f) { ax = 0.f; ay = 0.f; az = 0.f; }

    float sx, cx, sy, cy, sz, cz;
    sincosf(ax, &sx, &cx);
    sincosf(ay, &sy, &cy);
    sincosf(az, &sz, &cz);

    // R = Rz @ Ry @ Rx (row-major)
    float* R = out_R + (size_t)b * 9;
    R[0] = cz * cy;
    R[1] = cz * sy * sx - sz * cx;
    R[2] = cz * sy * cx + sz * sx;
    R[3] = sz * cy;
    R[4] = sz * sy * sx + cz * cx;
    R[5] = sz * sy * cx - cz * sx;
    R[6] = -sy;
    R[7] = cy * sx;
    R[8] = cy * cx;
}

// ---------------- Kernel 2: rotate points + quaternions --------------------
// 256 threads = 8 waves; each wave owns 32 points of one batch element.
// Per wave: 128 A-rows (3 rot-columns + pos per point, K padded to 4)
// x  B = R^T (padded 4x16)  via 8x V_WMMA_F32_16X16X4_F32.
__global__ __launch_bounds__(256) void
RotRandomizer_rotate_kernel(const float* __restrict__ pos,
                            const float* __restrict__ quat,
                            const float* __restrict__ Rg,
                            float* __restrict__ out_pos,
                            float* __restrict__ out_quat) {
    __shared__ float lds[8 * 1024];   // per wave: 512 f A-stage + 512 f D-stage

    const int tid  = threadIdx.x;
    const int wave = tid >> 5;
    const int lane = tid & 31;
    const int h    = lane & 15;
    const bool hiL = lane >= 16;
    const int b    = blockIdx.x * 2 + (wave >> 2);
    const int pt   = ((wave & 3) << 5) + lane;
    const int abase = wave * 1024;
    const int obase = abase + 512;

    const size_t idx = (size_t)b * TPTS + pt;

    // -------- load inputs (coalesced: b96 pos, b128 quat) --------
    const float* pi = pos + idx * 3;
    const float px = pi[0], py = pi[1], pz = pi[2];
    const float4 q = *((const float4*)quat + idx);   // xyzw
    const float qx = q.x, qy = q.y, qz = q.z, qw = q.w;

    // -------- quat -> rot (reference _quat_to_mat, wxyz order) --------
    const float two = 2.0f / (qw * qw + qx * qx + qy * qy + qz * qz);
    const float m00 = 1.0f - two * (qy * qy + qz * qz);
    const float m01 = two * (qx * qy - qz * qw);
    const float m02 = two * (qx * qz + qy * qw);
    const float m10 = two * (qx * qy + qz * qw);
    const float m11 = 1.0f - two * (qx * qx + qz * qz);
    const float m12 = two * (qy * qz - qx * qw);
    const float m20 = two * (qx * qz - qy * qw);
    const float m21 = two * (qy * qz + qx * qw);
    const float m22 = 1.0f - two * (qx * qx + qy * qy);

    // -------- stage A rows into LDS: row 4p+s = rot column s, row 4p+3 = pos
    float4* ap = (float4*)&lds[abase + lane * 16];
    ap[0] = make_float4(m00, m10, m20, 0.f);
    ap[1] = make_float4(m01, m11, m21, 0.f);
    ap[2] = make_float4(m02, m12, m22, 0.f);
    ap[3] = make_float4(px, py, pz, 0.f);

    // -------- B operand = R^T in WMMA 4x16 layout --------
    // VGPR0: lanes 0-15 -> B[0][n]=R[n][0]; lanes 16-31 -> B[2][n]=R[n][2]
    // VGPR1: lanes 0-15 -> B[1][n]=R[n][1]; lanes 16-31 -> B[3][n]=0
    const float* Rb = Rg + (size_t)b * 9;
    float b0v = 0.f, b1v = 0.f;
    if (h < 3) {
        if (!hiL) { b0v = Rb[h * 3 + 0]; b1v = Rb[h * 3 + 1]; }
        else      { b0v = Rb[h * 3 + 2]; }
    }
    v2f bm; bm.x = b0v; bm.y = b1v;

    __syncthreads();

    const int acol = hiL ? 2 : 0;   // A: lanes 16-31 hold K=2,3
    const int drow = hiL ? 8 : 0;   // D: lanes 16-31 hold M=j+8

    // -------- hoist all 8 A-operand LDS loads: they pipeline under one wait
    v2f am[8];
#pragma unroll
    for (int c = 0; c < 8; ++c) {
        const float* ar = &lds[abase + (c * 16 + h) * 4 + acol];
        am[c].x = ar[0];
        am[c].y = ar[1];
    }

    // -------- 8 back-to-back WMMAs cover the wave's 128 rows --------
    const v8f czero = {0.f, 0.f, 0.f, 0.f, 0.f, 0.f, 0.f, 0.f};
#pragma unroll
    for (int c = 0; c < 8; ++c) {
        v8f d = __builtin_amdgcn_wmma_f32_16x16x4_f32(
            false, am[c], false, bm, (short)0, czero, false, false);
        // only N=0..2 of D are meaningful: scatter them back to LDS
        if (h < 3) {
#pragma unroll
            for (int j = 0; j < 8; ++j) {
                lds[obase + (c * 16 + j + drow) * 4 + h] = d[j];
            }
        }
    }

    __syncthreads();

    // -------- gather this point's results: M columns + rotated pos --------
    const float4* op = (const float4*)&lds[obase + lane * 16];
    const float4 c0 = op[0];   // (M00, M10, M20, *)
    const float4 c1 = op[1];   // (M01, M11, M21, *)
    const float4 c2 = op[2];   // (M02, M12, M22, *)
    const float4 rp = op[3];   // rotated position

    const float M00 = c0.x, M10 = c0.y, M20 = c0.z;
    const float M01 = c1.x, M11 = c1.y, M21 = c1.z;
    const float M02 = c2.x, M12 = c2.y, M22 = c2.z;

    // -------- mat -> quat (reference _mat_to_quat) --------
    const float t0 = 1.0f + M00 + M11 + M22;
    const float t1 = 1.0f + M00 - M11 - M22;
    const float t2 = 1.0f - M00 + M11 - M22;
    const float t3 = 1.0f - M00 - M11 + M22;
    const float qa0 = sqrtf(fmaxf(t0, 1e-8f));
    const float qa1 = sqrtf(fmaxf(t1, 1e-8f));
    const float qa2 = sqrtf(fmaxf(t2, 1e-8f));
    const float qa3 = sqrtf(fmaxf(t3, 1e-8f));

    int best = 0; float bv = qa0;
    if (qa1 > bv) { best = 1; bv = qa1; }
    if (qa2 > bv) { best = 2; bv = qa2; }
    if (qa3 > bv) { best = 3; bv = qa3; }

    float W, X, Y, Z;
    if (best == 0)      { W = qa0 * qa0; X = M21 - M12; Y = M02 - M20; Z = M10 - M01; }
    else if (best == 1) { W = M21 - M12; X = qa1 * qa1; Y = M01 + M10; Z = M02 + M20; }
    else if (best == 2) { W = M02 - M20; X = M10 + M01; Y = qa2 * qa2; Z = M12 + M21; }
    else                { W = M10 - M01; X = M20 + M02; Y = M21 + M12; Z = qa3 * qa3; }
    const float inv = 1.0f / (2.0f * fmaxf(bv, 0.1f));
    W *= inv; X *= inv; Y *= inv; Z *= inv;

    // -------- stores (coalesced) --------
    float* po = out_pos + idx * 3;
    po[0] = rp.x; po[1] = rp.y; po[2] = rp.z;
    ((float4*)out_quat)[idx] = make_float4(X, Y, Z, W);  // back to xyzw
}

// ---------------------------------------------------------------------------
extern "C" void kernel_launch(void* const* d_in, const int* in_sizes, int n_in,
                              void* d_out, int out_size, void* d_ws, size_t ws_size,
                              hipStream_t stream) {
    const float* pos  = (const float*)d_in[0];   // [B,T,3]
    const float* quat = (const float*)d_in[1];   // [B,T,4]

    float* out      = (float*)d_out;
    float* out_pos  = out;                                          // B*T*3
    float* out_quat = out + (size_t)BATCH * TPTS * 3;               // B*T*4
    float* out_R    = out + (size_t)BATCH * TPTS * 3
                          + (size_t)BATCH * TPTS * 4;               // B*9

    RotRandomizer_build_R_kernel<<<BATCH / 256, 256, 0, stream>>>(out_R);
    RotRandomizer_rotate_kernel<<<BATCH / 2, 256, 0, stream>>>(
        pos, quat, out_R, out_pos, out_quat);
}